// GCN_3530463118095
// MI455X (gfx1250) — compile-verified
//
#include <hip/hip_runtime.h>

typedef __attribute__((ext_vector_type(2))) float v2f;
typedef __attribute__((ext_vector_type(8))) float v8f;

#define F0 512
#define F1 256
#define F2 64

// ---------------- degree accumulation (segment_sum of ones) ----------------
__global__ void gcn_degree_kernel(const int* __restrict__ src, const int* __restrict__ dst,
                                  float* __restrict__ deg_out, float* __restrict__ deg_in, int E) {
    int e = blockIdx.x * blockDim.x + threadIdx.x;
    if (e < E) {
        atomicAdd(&deg_out[src[e]], 1.0f);
        atomicAdd(&deg_in[dst[e]], 1.0f);
    }
}

// ---------------- deg -> norm (in place): deg>0 ? rsqrt(max(deg,1)) : 1 ----
__global__ void gcn_norm_kernel(float* __restrict__ a, float* __restrict__ b, int n) {
    int i = blockIdx.x * blockDim.x + threadIdx.x;
    if (i < n) {
        float da = a[i];
        a[i] = (da > 0.0f) ? rsqrtf(fmaxf(da, 1.0f)) : 1.0f;
        float db = b[i];
        b[i] = (db > 0.0f) ? rsqrtf(fmaxf(db, 1.0f)) : 1.0f;
    }
}

// ---------------- fused (rowscale ∘ GEMM) via f32 WMMA ---------------------
// C[m,n] = sum_k (A[m,k]*normA[m]) * B[k,n]
// One wave computes a 16x64 strip of C: 4 accumulators, A fragment reused 4x.
// A frag layout (V_WMMA_F32_16X16X4_F32): lanes 0-15 rows, vgpr0/1 = K,K+1;
// lanes 16-31 rows, vgpr0/1 = K+2,K+3  -> one contiguous float2 per lane.
__global__ __launch_bounds__(256) void gcn_gemm_wmma_f32(
    const float* __restrict__ A, const float* __restrict__ B,
    const float* __restrict__ normA, float* __restrict__ C,
    int M, int K, int N) {
    int wave = (blockIdx.x * blockDim.x + threadIdx.x) >> 5;
    int lane = threadIdx.x & 31;
    int tilesN = N >> 6;                 // 64 columns per wave
    int tm = wave / tilesN;
    int tn = wave % tilesN;
    int m0 = tm << 4;
    int n0 = tn << 6;
    if (m0 >= M) return;

    int half = lane >> 4;                // 0: K,K+1   1: K+2,K+3
    int l    = lane & 15;
    int row  = m0 + l;
    float scale = normA[row];
    const float* Arow = A + (size_t)row * K;

    v8f acc0 = {}, acc1 = {}, acc2 = {}, acc3 = {};
    for (int k0 = 0; k0 < K; k0 += 4) {
        int ka = k0 + (half << 1);
        float2 at = *(const float2*)(Arow + ka);
        v2f a;
        a.x = at.x * scale;
        a.y = at.y * scale;

        const float* Bp = B + (size_t)ka * N + n0 + l;   // B[ka][n0+l], next row +N
        v2f b0, b1, b2, b3;
        b0.x = Bp[0];   b0.y = Bp[N];
        b1.x = Bp[16];  b1.y = Bp[N + 16];
        b2.x = Bp[32];  b2.y = Bp[N + 32];
        b3.x = Bp[48];  b3.y = Bp[N + 48];

        acc0 = __builtin_amdgcn_wmma_f32_16x16x4_f32(false, a, false, b0, (short)0, acc0, false, false);
        acc1 = __builtin_amdgcn_wmma_f32_16x16x4_f32(false, a, false, b1, (short)0, acc1, false, false);
        acc2 = __builtin_amdgcn_wmma_f32_16x16x4_f32(false, a, false, b2, (short)0, acc2, false, false);
        acc3 = __builtin_amdgcn_wmma_f32_16x16x4_f32(false, a, false, b3, (short)0, acc3, false, false);
    }

    // C/D layout: lanes 0-15 -> M = m0+i, lanes 16-31 -> M = m0+8+i; col = n0 + 16*t + l
    #pragma unroll
    for (int i = 0; i < 8; ++i) {
        int r = m0 + i + (half << 3);
        size_t base = (size_t)r * N + n0 + l;
        C[base]      = acc0[i];
        C[base + 16] = acc1[i];
        C[base + 32] = acc2[i];
        C[base + 48] = acc3[i];
    }
}

// ---------------- edge gather + scatter-add: AGG[dst] += H[src] ------------
// thread = (edge, 4-float chunk); lanes cover contiguous chunks -> coalesced.
__global__ void gcn_scatter_add_kernel(const float* __restrict__ H, const int* __restrict__ src,
                                       const int* __restrict__ dst, float* __restrict__ AGG,
                                       int E, int F) {
    long tid = (long)blockIdx.x * blockDim.x + threadIdx.x;
    int cpe = F >> 2;
    long e = tid / cpe;
    int  c = (int)(tid % cpe) << 2;
    if (e >= E) return;
    int s = src[e];
    int d = dst[e];
    float4 v = *(const float4*)(H + (size_t)s * F + c);
    float* o = AGG + (size_t)d * F + c;
    atomicAdd(o + 0, v.x);
    atomicAdd(o + 1, v.y);
    atomicAdd(o + 2, v.z);
    atomicAdd(o + 3, v.w);
}

// ---------------- OUT = act(AGG * norm_dst[node] + bias[f]) ----------------
__global__ void gcn_bias_act_kernel(const float* __restrict__ AGG, const float* __restrict__ norm_dst,
                                    const float* __restrict__ bias, float* __restrict__ OUT,
                                    long total, int F, int relu) {
    long i = (long)blockIdx.x * blockDim.x + threadIdx.x;
    if (i < total) {
        int node = (int)(i / F);
        int f    = (int)(i % F);
        float v = AGG[i] * norm_dst[node] + bias[f];
        OUT[i] = relu ? fmaxf(v, 0.0f) : v;
    }
}

extern "C" void kernel_launch(void* const* d_in, const int* in_sizes, int n_in,
                              void* d_out, int out_size, void* d_ws, size_t ws_size,
                              hipStream_t stream) {
    const float* feat = (const float*)d_in[0];
    const int*   src  = (const int*)d_in[1];
    const int*   dst  = (const int*)d_in[2];
    const float* W1   = (const float*)d_in[3];
    const float* b1   = (const float*)d_in[4];
    const float* W2   = (const float*)d_in[5];
    const float* b2   = (const float*)d_in[6];
    float* out = (float*)d_out;

    const int N = in_sizes[0] / F0;   // 50000
    const int E = in_sizes[1];        // 800000

    float* ws       = (float*)d_ws;
    float* norm_src = ws;                               // N
    float* norm_dst = norm_src + N;                     // N
    float* H1       = norm_dst + N;                     // N*F1
    float* AGG1     = H1   + (size_t)N * F1;            // N*F1 (becomes relu'd H1b)
    float* H2       = AGG1 + (size_t)N * F1;            // N*F2

    // zero-init accumulators (capture-legal async memsets)
    hipMemsetAsync(norm_src, 0, (size_t)2 * N * sizeof(float), stream);
    hipMemsetAsync(AGG1, 0, (size_t)N * F1 * sizeof(float), stream);
    hipMemsetAsync(out,  0, (size_t)N * F2 * sizeof(float), stream);

    // degrees -> norms
    gcn_degree_kernel<<<(E + 255) / 256, 256, 0, stream>>>(src, dst, norm_src, norm_dst, E);
    gcn_norm_kernel<<<(N + 255) / 256, 256, 0, stream>>>(norm_src, norm_dst, N);

    // layer 1: H1 = (feat * norm_src) @ W1
    {
        int waves  = (N / 16) * (F1 / 64);
        int blocks = (waves * 32 + 255) / 256;
        gcn_gemm_wmma_f32<<<blocks, 256, 0, stream>>>(feat, W1, norm_src, H1, N, F0, F1);
    }
    // AGG1[dst] += H1[src]
    {
        long threads = (long)E * (F1 / 4);
        gcn_scatter_add_kernel<<<(int)((threads + 255) / 256), 256, 0, stream>>>(H1, src, dst, AGG1, E, F1);
    }
    // H1b = relu(AGG1 * norm_dst + b1)   (in place over AGG1)
    {
        long total = (long)N * F1;
        gcn_bias_act_kernel<<<(int)((total + 255) / 256), 256, 0, stream>>>(AGG1, norm_dst, b1, AGG1, total, F1, 1);
    }
    // layer 2: H2 = (H1b * norm_src) @ W2
    {
        int waves  = (N / 16) * (F2 / 64);
        int blocks = (waves * 32 + 255) / 256;
        gcn_gemm_wmma_f32<<<blocks, 256, 0, stream>>>(AGG1, W2, norm_src, H2, N, F1, F2);
    }
    // out[dst] += H2[src]
    {
        long threads = (long)E * (F2 / 4);
        gcn_scatter_add_kernel<<<(int)((threads + 255) / 256), 256, 0, stream>>>(H2, src, dst, out, E, F2);
    }
    // out = out * norm_dst + b2
    {
        long total = (long)N * F2;
        gcn_bias_act_kernel<<<(int)((total + 255) / 256), 256, 0, stream>>>(out, norm_dst, b2, out, total, F2, 0);
    }
}